// GraphConvolution_46634754900286
// MI455X (gfx1250) — compile-verified
//
#include <hip/hip_runtime.h>
#include <hip/hip_bf16.h>
#include <stdint.h>

// GCN layer: out[b,n,o] = dhat[n] * sum_m adj[n,m] * dhat[m] * (input[b,m,:] @ W)[o]
// N=8192, B=4, F=128. Main GEMM: adj[8192x8192] @ S[8192x512] via V_WMMA_F32_16X16X4_F32,
// with double-buffered GLOBAL_LOAD_ASYNC_TO_LDS_B128 staging (ASYNCcnt pipeline).

#define NN   8192
#define BB   4
#define FD   128
#define COLS (BB * FD)   // 512

typedef __attribute__((ext_vector_type(2))) float v2f;
typedef __attribute__((ext_vector_type(4))) float v4f;
typedef __attribute__((ext_vector_type(8))) float v8f;

// CDNA5 async memory->LDS copy (ISA 15.18.3 op 98). INST_OFFSET adds to both
// the global address and the LDS destination address (ISA 08 §4.4).
#define ASYNC_CP_B128(ldsoff, gptr, imm)                                     \
  asm volatile("global_load_async_to_lds_b128 %0, %1, off offset:" imm       \
               :: "v"(ldsoff), "v"(gptr) : "memory")
#define WAIT_ASYNCCNT0() asm volatile("s_wait_asynccnt 0x0" ::: "memory")

// Low 32 bits of a generic LDS-aperture address = wave-relative LDS byte
// offset (ISA 00 §10.2 aperture mapping).
__device__ inline uint32_t lds_off(const void* p) {
  return (uint32_t)(uintptr_t)p;
}

// ---------------------------------------------------------------------------
// Kernel 1: column degrees + rsqrt.  deg[j] = sum_i adj[i,j]; dhat[j]=rsqrt.
// ---------------------------------------------------------------------------
__global__ void gcn_degree_kernel(const float* __restrict__ adj,
                                  float* __restrict__ dhat) {
  const int j = blockIdx.x * blockDim.x + threadIdx.x;   // 0..8191, coalesced
  float s0 = 0.f, s1 = 0.f, s2 = 0.f, s3 = 0.f;
  for (int i = 0; i < NN; i += 4) {
    s0 += adj[(size_t)(i + 0) * NN + j];
    s1 += adj[(size_t)(i + 1) * NN + j];
    s2 += adj[(size_t)(i + 2) * NN + j];
    s3 += adj[(size_t)(i + 3) * NN + j];
  }
  dhat[j] = rsqrtf((s0 + s1) + (s2 + s3));
}

// ---------------------------------------------------------------------------
// Kernel 2: S[m, b*128+o] = dhat[m] * sum_f input[b,m,f] * W[f,o]
// ---------------------------------------------------------------------------
__global__ void gcn_support_kernel(const float* __restrict__ input,
                                   const float* __restrict__ weight,
                                   const float* __restrict__ dhat,
                                   float* __restrict__ S) {
  __shared__ float w_lds[64 * FD];                       // 32 KB
  const int tid = threadIdx.x;                           // 0..255
  const int b   = blockIdx.y;
  const int m   = blockIdx.x * 16 + (tid >> 4);
  const int o0  = (tid & 15) * 8;
  const float* in_row = input + ((size_t)b * NN + m) * FD;

  float acc[8] = {};
  for (int fc = 0; fc < FD; fc += 64) {
    __syncthreads();
    for (int i = tid * 4; i < 64 * FD; i += 256 * 4)
      *(v4f*)&w_lds[i] = *(const v4f*)&weight[fc * FD + i];
    __syncthreads();
#pragma unroll 8
    for (int f = 0; f < 64; ++f) {
      const float iv = in_row[fc + f];
      const float* wr = &w_lds[f * FD + o0];
#pragma unroll
      for (int k = 0; k < 8; ++k) acc[k] += iv * wr[k];
    }
  }
  const float sc = dhat[m];
  float* outp = S + (size_t)m * COLS + b * FD + o0;
#pragma unroll
  for (int k = 0; k < 8; ++k) outp[k] = acc[k] * sc;
}

// ---------------------------------------------------------------------------
// Kernel 3: C = adj @ S with dhat[n] row-scale epilogue, out[b,n,o].
// 256 thr = 8 waves; block tile 128x128 (one batch per column block);
// wave tile 64x32 = 4x2 WMMA accumulators. Double-buffered async LDS staging.
// ---------------------------------------------------------------------------
#define KT   32
#define ASTR 36    // padded LDS stride for A (floats)
#define BSTR 132   // padded LDS stride for B (floats)

__global__ void __launch_bounds__(256)
gcn_aggregate_kernel(const float* __restrict__ adj,
                     const float* __restrict__ S,
                     const float* __restrict__ dhat,
                     float* __restrict__ out) {
  __shared__ float As[2][128 * ASTR];   // 2 x 18 KB   [row][k]
  __shared__ float Bs[2][KT * BSTR];    // 2 x 16.5 KB [k][col]
  __shared__ float dh[128];

  const int tid    = threadIdx.x;        // 0..255
  const int lane   = tid & 31;
  const int wave   = tid >> 5;           // 0..7
  const int lane15 = lane & 15;
  const int hi     = lane >> 4;          // 0: K={0,1}/M=r ; 1: K={2,3}/M=r+8
  const int rowBase = blockIdx.x * 128;
  const int cb      = blockIdx.y;        // column block == batch index

  if (tid < 128) dh[tid] = dhat[rowBase + tid];

  const int wRow = (wave >> 2) * 64;     // 0 / 64
  const int wCol = (wave & 3) * 32;      // 0 / 32 / 64 / 96

  v8f zero = {};
  v8f acc[4][2];
#pragma unroll
  for (int i = 0; i < 4; ++i)
#pragma unroll
    for (int j = 0; j < 2; ++j) acc[i][j] = zero;

  // Per-thread staging assignment (64 B of A + 64 B of B per thread per tile)
  const int arow_r = tid >> 1;                    // 0..127 (A row)
  const int acol0  = (tid & 1) * 16;              // 0 / 16
  const float* aptr = adj + (size_t)(rowBase + arow_r) * NN + acol0;
  const int bk  = tid >> 3;                       // 0..31 (B k-row)
  const int bc0 = (tid & 7) * 16;                 // 0..112
  const float* sptr = S + (size_t)bk * COLS + cb * FD + bc0;

  uint32_t aoff[2], boff[2];
#pragma unroll
  for (int u = 0; u < 2; ++u) {
    aoff[u] = lds_off(&As[u][arow_r * ASTR + acol0]);
    boff[u] = lds_off(&Bs[u][bk * BSTR + bc0]);
  }

  // prefetch tile 0 into buffer 0
  {
    const float* ga = aptr;
    const float* gb = sptr;
    ASYNC_CP_B128(aoff[0], ga, "0");
    ASYNC_CP_B128(aoff[0], ga, "16");
    ASYNC_CP_B128(aoff[0], ga, "32");
    ASYNC_CP_B128(aoff[0], ga, "48");
    ASYNC_CP_B128(boff[0], gb, "0");
    ASYNC_CP_B128(boff[0], gb, "16");
    ASYNC_CP_B128(boff[0], gb, "32");
    ASYNC_CP_B128(boff[0], gb, "48");
  }

  int cur = 0;
  for (int kt = 0; kt < NN; kt += KT, cur ^= 1) {
    WAIT_ASYNCCNT0();            // this wave's prefetch for buf[cur] landed
    __syncthreads();             // everyone's landed; buf[cur^1] compute done

    if (kt + KT < NN) {          // prefetch next tile into the other buffer
      const int nxt = cur ^ 1;
      const float* ga = aptr + (kt + KT);
      const float* gb = sptr + (size_t)(kt + KT) * COLS;
      ASYNC_CP_B128(aoff[nxt], ga, "0");
      ASYNC_CP_B128(aoff[nxt], ga, "16");
      ASYNC_CP_B128(aoff[nxt], ga, "32");
      ASYNC_CP_B128(aoff[nxt], ga, "48");
      ASYNC_CP_B128(boff[nxt], gb, "0");
      ASYNC_CP_B128(boff[nxt], gb, "16");
      ASYNC_CP_B128(boff[nxt], gb, "32");
      ASYNC_CP_B128(boff[nxt], gb, "48");
    }

    const float* Asc = &As[cur][0];
    const float* Bsc = &Bs[cur][0];
#pragma unroll
    for (int kk = 0; kk < KT; kk += 4) {
      // A frag: 16x4 f32 — lane m = lane15, VGPR0/1 = K {0,1} (+2 if hi)
      v2f a[4];
#pragma unroll
      for (int mi = 0; mi < 4; ++mi)
        a[mi] = *(const v2f*)&Asc[(wRow + mi * 16 + lane15) * ASTR + kk + 2 * hi];
      // B frag: 4x16 f32 — VGPR r: lanes0-15 K=r, lanes16-31 K=r+2
      v2f b[2];
#pragma unroll
      for (int ni = 0; ni < 2; ++ni) {
        const int c = wCol + ni * 16 + lane15;
        b[ni].x = Bsc[(kk + 2 * hi + 0) * BSTR + c];
        b[ni].y = Bsc[(kk + 2 * hi + 1) * BSTR + c];
      }
#pragma unroll
      for (int mi = 0; mi < 4; ++mi)
#pragma unroll
        for (int ni = 0; ni < 2; ++ni)
          acc[mi][ni] = __builtin_amdgcn_wmma_f32_16x16x4_f32(
              false, a[mi], false, b[ni], (short)0, acc[mi][ni], false, false);
    }
  }

  // Epilogue: C/D layout VGPR r: lanes0-15 -> M=r, lanes16-31 -> M=r+8
#pragma unroll
  for (int mi = 0; mi < 4; ++mi) {
#pragma unroll
    for (int ni = 0; ni < 2; ++ni) {
#pragma unroll
      for (int r = 0; r < 8; ++r) {
        const int ml = wRow + mi * 16 + r + 8 * hi;   // local row 0..127
        const int o  = wCol + ni * 16 + lane15;       // feature 0..127
        const float v = acc[mi][ni][r] * dh[ml];
        out[((size_t)cb * NN + (rowBase + ml)) * FD + o] = v;
      }
    }
  }
}

// ---------------------------------------------------------------------------
extern "C" void kernel_launch(void* const* d_in, const int* in_sizes, int n_in,
                              void* d_out, int out_size, void* d_ws, size_t ws_size,
                              hipStream_t stream) {
  (void)in_sizes; (void)n_in; (void)out_size; (void)ws_size;
  const float* input  = (const float*)d_in[0];   // [B,N,F]
  const float* adj    = (const float*)d_in[1];   // [N,N]
  const float* weight = (const float*)d_in[2];   // [F,F]
  float* out = (float*)d_out;                    // [B,N,F]

  float* ws   = (float*)d_ws;
  float* dhat = ws;            // 8192 floats
  float* S    = ws + NN;       // 8192*512 floats (~16.8 MB total ws use)

  gcn_degree_kernel<<<NN / 256, 256, 0, stream>>>(adj, dhat);
  gcn_support_kernel<<<dim3(NN / 16, BB), 256, 0, stream>>>(input, weight, dhat, S);
  gcn_aggregate_kernel<<<dim3(NN / 128, BB), 256, 0, stream>>>(adj, S, dhat, out);
}